// LightGCN_44753559225051
// MI455X (gfx1250) — compile-verified
//
#include <hip/hip_runtime.h>
#include <cstdint>
#include <cstddef>

// ---------------------------------------------------------------------------
// LightGCN propagation on MI455X (gfx1250).
// Memory/atomic-bound COO SpMM: edge streams staged to LDS with the CDNA5
// async global->LDS DMA path (ASYNCcnt / s_wait_asynccnt), gathers via
// global_load_b128, scatters via global_atomic_add_f32 (resolve in 192MB L2).
// ---------------------------------------------------------------------------

static constexpr int       kUsers   = 100000;
static constexpr int       kItems   = 50000;
static constexpr int       kNodes   = kUsers + kItems;   // 150000
static constexpr int       kDim     = 64;
static constexpr long long kEdges   = 5000000;

static constexpr int       kBlock         = 256;          // 8 waves (wave32)
static constexpr int       kWavesPerBlock = kBlock / 32;
static constexpr int       kTileEdges     = 128;          // per wave tile
static constexpr long long kFullTiles     = kEdges / kTileEdges;      // 39062
static constexpr long long kTailStart     = kFullTiles * kTileEdges;  // 4999936

// ---- CDNA5 async global->LDS DMA helpers (VFLAT async encoding) -----------
__device__ __forceinline__ void async_load_b128(uint32_t lds_off, const void* gaddr) {
#if defined(__gfx1250__)
  // vdst = LDS byte address (VGPR), vaddr = 64-bit global address, no SADDR.
  asm volatile("global_load_async_to_lds_b128 %0, %1, off"
               : : "v"(lds_off), "v"(gaddr) : "memory");
#else
  (void)lds_off; (void)gaddr;
#endif
}

__device__ __forceinline__ void wait_async_all() {
#if defined(__gfx1250__)
  asm volatile("s_wait_asynccnt 0x0" : : : "memory");
#endif
}

__device__ __forceinline__ void wait_async_le3() {
#if defined(__gfx1250__)
  asm volatile("s_wait_asynccnt 0x3" : : : "memory");
#endif
}

// Low 32 bits of a generic pointer into LDS == LDS byte offset (ISA aperture
// rule: the LDS address is the pointer's low 32 bits).
__device__ __forceinline__ uint32_t lds_u32(const void* p) {
  return (uint32_t)(uintptr_t)p;
}

// ---------------------------------------------------------------------------
// acc = bufA = concat(user_w, item_w); bufB = 0
__global__ __launch_bounds__(kBlock)
void lightgcn_init(const float* __restrict__ user_w,
                   const float* __restrict__ item_w,
                   float* __restrict__ bufA,
                   float* __restrict__ bufB,
                   float* __restrict__ acc) {
  const size_t n4 = (size_t)kNodes * kDim / 4;
  const size_t u4 = (size_t)kUsers * kDim / 4;
  for (size_t i = (size_t)blockIdx.x * blockDim.x + threadIdx.x; i < n4;
       i += (size_t)gridDim.x * blockDim.x) {
    float4 v = (i < u4) ? reinterpret_cast<const float4*>(user_w)[i]
                        : reinterpret_cast<const float4*>(item_w)[i - u4];
    reinterpret_cast<float4*>(bufA)[i] = v;
    reinterpret_cast<float4*>(acc)[i]  = v;
    reinterpret_cast<float4*>(bufB)[i] = make_float4(0.f, 0.f, 0.f, 0.f);
  }
}

// acc = (acc + layer) * scale ; zero_buf = 0 (preps next layer's destination)
__global__ __launch_bounds__(kBlock)
void lightgcn_accum_scale_zero(float* __restrict__ acc,
                               const float* __restrict__ layer,
                               float* __restrict__ zero_buf,
                               float scale) {
  const size_t n4 = (size_t)kNodes * kDim / 4;
  for (size_t i = (size_t)blockIdx.x * blockDim.x + threadIdx.x; i < n4;
       i += (size_t)gridDim.x * blockDim.x) {
    float4 a = reinterpret_cast<float4*>(acc)[i];
    float4 l = reinterpret_cast<const float4*>(layer)[i];
    a.x = (a.x + l.x) * scale;
    a.y = (a.y + l.y) * scale;
    a.z = (a.z + l.z) * scale;
    a.w = (a.w + l.w) * scale;
    reinterpret_cast<float4*>(acc)[i] = a;
    reinterpret_cast<float4*>(zero_buf)[i] = make_float4(0.f, 0.f, 0.f, 0.f);
  }
}

// dst[r] += v_e * src[c_e]  (dst pre-zeroed). Edge tuples double-buffered in
// per-wave LDS via async DMA; 16 lanes per edge, float4 per lane.
__global__ __launch_bounds__(kBlock)
void lightgcn_spmm(const int* __restrict__ rows,
                   const int* __restrict__ cols,
                   const float* __restrict__ vals,
                   const float* __restrict__ src,
                   float* __restrict__ dst) {
  __shared__ int   s_rows[kWavesPerBlock][2][kTileEdges];
  __shared__ int   s_cols[kWavesPerBlock][2][kTileEdges];
  __shared__ float s_vals[kWavesPerBlock][2][kTileEdges];

  const int tid   = threadIdx.x;
  const int w     = tid >> 5;
  const int lane  = tid & 31;
  const int sub   = lane & 15;   // 0..15: which float4 of the 64-dim row
  const int epair = lane >> 4;   // 0..1 : which edge of the current pair

  const long long gw     = (long long)blockIdx.x * kWavesPerBlock + w;
  const long long stride = (long long)gridDim.x * kWavesPerBlock;

  const uint32_t laneOff = (uint32_t)lane * 16u;   // 16 B (4 edges) per lane

  long long t  = gw;
  int      buf = 0;
  if (t < kFullTiles) {
    const long long eb = t * kTileEdges + (long long)lane * 4;
    async_load_b128(lds_u32(&s_rows[w][0][0]) + laneOff, rows + eb);
    async_load_b128(lds_u32(&s_cols[w][0][0]) + laneOff, cols + eb);
    async_load_b128(lds_u32(&s_vals[w][0][0]) + laneOff, vals + eb);
  }

  for (; t < kFullTiles; t += stride) {
    const long long nt = t + stride;
    if (nt < kFullTiles) {
      // Prefetch next tile into the other buffer, then wait for the 3 async
      // loads of the *current* tile (async loads complete in order).
      const int nb = buf ^ 1;
      const long long eb = nt * kTileEdges + (long long)lane * 4;
      async_load_b128(lds_u32(&s_rows[w][nb][0]) + laneOff, rows + eb);
      async_load_b128(lds_u32(&s_cols[w][nb][0]) + laneOff, cols + eb);
      async_load_b128(lds_u32(&s_vals[w][nb][0]) + laneOff, vals + eb);
      wait_async_le3();
    } else {
      wait_async_all();
    }

#pragma unroll 4
    for (int i = 0; i < kTileEdges / 2; ++i) {
      const int   e = 2 * i + epair;
      const int   r = s_rows[w][buf][e];
      const int   c = s_cols[w][buf][e];
      const float v = s_vals[w][buf][e];
      const float4 x =
          *reinterpret_cast<const float4*>(src + ((size_t)c * kDim + sub * 4));
      float* d = dst + ((size_t)r * kDim + sub * 4);
      atomicAdd(d + 0, v * x.x);
      atomicAdd(d + 1, v * x.y);
      atomicAdd(d + 2, v * x.z);
      atomicAdd(d + 3, v * x.w);
    }
    buf ^= 1;
  }

  // Tail edges (kEdges % kTileEdges == 64): plain path, 16 lanes per edge.
  const long long g16 = ((long long)blockIdx.x * kBlock + tid) >> 4;
  const long long n16 = ((long long)gridDim.x * kBlock) >> 4;
  for (long long e = kTailStart + g16; e < kEdges; e += n16) {
    const int   r = rows[e];
    const int   c = cols[e];
    const float v = vals[e];
    const float4 x =
        *reinterpret_cast<const float4*>(src + ((size_t)c * kDim + sub * 4));
    float* d = dst + ((size_t)r * kDim + sub * 4);
    atomicAdd(d + 0, v * x.x);
    atomicAdd(d + 1, v * x.y);
    atomicAdd(d + 2, v * x.z);
    atomicAdd(d + 3, v * x.w);
  }
}

// ---------------------------------------------------------------------------
extern "C" void kernel_launch(void* const* d_in, const int* in_sizes, int n_in,
                              void* d_out, int out_size, void* d_ws, size_t ws_size,
                              hipStream_t stream) {
  (void)in_sizes; (void)n_in; (void)out_size; (void)ws_size;

  const float* user_w = (const float*)d_in[0];
  const float* item_w = (const float*)d_in[1];
  const int*   rows   = (const int*)d_in[2];
  const int*   cols   = (const int*)d_in[3];
  const float* vals   = (const float*)d_in[4];

  float* acc  = (float*)d_out;                       // running sum (output)
  float* bufA = (float*)d_ws;                        // ping
  float* bufB = bufA + (size_t)kNodes * kDim;        // pong

  const size_t n4       = (size_t)kNodes * kDim / 4; // 2.4M float4
  const int    ewBlocks = (int)((n4 + kBlock - 1) / kBlock);   // 9375
  const int    spBlocks = 1536;                      // 12288 waves over 39062 tiles

  lightgcn_init<<<ewBlocks, kBlock, 0, stream>>>(user_w, item_w, bufA, bufB, acc);

  // Layer 1: B = A*Adj ; acc += B ; zero A
  lightgcn_spmm<<<spBlocks, kBlock, 0, stream>>>(rows, cols, vals, bufA, bufB);
  lightgcn_accum_scale_zero<<<ewBlocks, kBlock, 0, stream>>>(acc, bufB, bufA, 1.0f);

  // Layer 2: A = B*Adj ; acc += A ; zero B
  lightgcn_spmm<<<spBlocks, kBlock, 0, stream>>>(rows, cols, vals, bufB, bufA);
  lightgcn_accum_scale_zero<<<ewBlocks, kBlock, 0, stream>>>(acc, bufA, bufB, 1.0f);

  // Layer 3: B = A*Adj ; acc = (acc + B) / 4 ; (zeroing A is harmless)
  lightgcn_spmm<<<spBlocks, kBlock, 0, stream>>>(rows, cols, vals, bufA, bufB);
  lightgcn_accum_scale_zero<<<ewBlocks, kBlock, 0, stream>>>(acc, bufB, bufA, 0.25f);
}